// GIN_82377472737540
// MI455X (gfx1250) — compile-verified
//
#include <hip/hip_runtime.h>
#include <math.h>

// ---------------------------------------------------------------------------
// GIN forward for MI455X (gfx1250, wave32).
//
// Roofline: ~5.6 GB traffic/call vs ~20 GFLOP -> bandwidth/atomic bound.
// GEMMs therefore use the *exact fp32* matrix path V_WMMA_F32_16X16X4_F32
// (no quantization risk, WMMA throughput is not the limiter).
//
// Input flattening assumption (setup_inputs() dict insertion order):
//   d_in[0]                 x            [100000,128] f32
//   d_in[1+13*L+0..12]      layer L:     eps, w1, b1, bn1_g, bn1_b, bn1_m,
//                                        bn1_v, w2, b2, bn_g, bn_b, bn_m, bn_v
//   d_in[40..43]            lin1_w, lin1_b, lin2_w, lin2_b
//   d_in[44]                edge_index   [2,1600000] i32
//   d_in[45]                batch        [100000] i32
//
// Workspace layout (floats), requires ws_size >= ~105 MB:
//   bufA[12.8M] | bufB[12.8M] | G[1008*128] | G2[1008*128] | packW[7*16384]
// ---------------------------------------------------------------------------

#define N_NODES   100000
#define N_EDGES   1600000
#define D_HID     128
#define N_GRAPHS  1000
#define N_CLASSES 10
#define GPAD      1008        // 63 * 16 (16-padded graph count)
#define BN_EPS    1e-5f

typedef __attribute__((ext_vector_type(2))) float v2f;
typedef __attribute__((ext_vector_type(8))) float v8f;

// ---------------------------------------------------------------------------
// tmp[i] = (1 + eps) * in[i]   (float4 vectorized; n4 = elems/4)
// ---------------------------------------------------------------------------
__global__ void gin_scale_copy(const float* __restrict__ in,
                               const float* __restrict__ eps,
                               float* __restrict__ out, int n4) {
  int i = blockIdx.x * blockDim.x + threadIdx.x;
  if (i >= n4) return;
  float s = 1.0f + eps[0];
  float4 v = reinterpret_cast<const float4*>(in)[i];
  v.x *= s; v.y *= s; v.z *= s; v.w *= s;
  reinterpret_cast<float4*>(out)[i] = v;
}

// ---------------------------------------------------------------------------
// Edge scatter-add: one wave32 per edge. Lane l loads x[src][4l..4l+3] as a
// float4 (512B coalesced row read per wave) and atomically adds into out[dst].
// ---------------------------------------------------------------------------
__global__ void gin_scatter_add(const float* __restrict__ x,
                                const int* __restrict__ src,
                                const int* __restrict__ dst,
                                float* __restrict__ out, int n_edges) {
  int t = blockIdx.x * blockDim.x + threadIdx.x;
  int e = t >> 5;
  int lane = t & 31;
  if (e >= n_edges) return;
  int s = src[e];
  int d = dst[e];
  float4 v = reinterpret_cast<const float4*>(x + (size_t)s * D_HID)[lane];
  float* o = out + (size_t)d * D_HID + lane * 4;
  atomicAdd(o + 0, v.x);
  atomicAdd(o + 1, v.y);
  atomicAdd(o + 2, v.z);
  atomicAdd(o + 3, v.w);
}

// ---------------------------------------------------------------------------
// Repack a row-major [128,128] weight into B-fragment order for
// v_wmma_f32_16x16x4_f32:
//   packed[((ntile*32 + kstep)*32 + lane)] = { W[k][n], W[k+1][n] }  (float2)
//   with n = ntile*16 + (lane&15), k = kstep*4 + 2*(lane>>4)
// so the GEMM's per-kstep B load is one fully coalesced float2 per lane.
// ---------------------------------------------------------------------------
__global__ void gin_pack_w(const float* __restrict__ W,
                           float* __restrict__ Wp) {
  int t = blockIdx.x * blockDim.x + threadIdx.x;   // 8192 threads total
  if (t >= 8 * 32 * 32) return;
  int lane  = t & 31;
  int kstep = (t >> 5) & 31;
  int ntile = t >> 10;
  int n = ntile * 16 + (lane & 15);
  int k = kstep * 4 + ((lane >> 4) << 1);
  float2 v = make_float2(W[k * D_HID + n], W[(k + 1) * D_HID + n]);
  reinterpret_cast<float2*>(Wp)[t] = v;
}

// ---------------------------------------------------------------------------
// out = relu( BN( A @ W + bias ) ), A:[M,128], W packed, M multiple of 16.
// block = 256 thr = 8 waves; each wave owns one 16-col N-tile of a 16-row
// M-tile; 32 chained v_wmma_f32_16x16x4_f32 over K=128.
// BN folded to scale/shift; bn_g==nullptr -> identity BN (plain bias+relu).
// A tile staged in LDS with stride 130 to avoid bank conflicts.
// ---------------------------------------------------------------------------
__global__ void gin_gemm_bn_relu(const float* __restrict__ A,
                                 const float* __restrict__ Wp,
                                 const float* __restrict__ bias,
                                 const float* __restrict__ bn_g,
                                 const float* __restrict__ bn_b,
                                 const float* __restrict__ bn_m,
                                 const float* __restrict__ bn_v,
                                 float* __restrict__ out) {
  __shared__ float As[16 * 130];
  const int tid = threadIdx.x;
  const int m0  = blockIdx.x * 16;

  // Stage 16x128 A tile: 8 floats per thread (2x float4 global, 4x float2 LDS).
  {
    int base = tid * 8;
    int r = base >> 7;
    int c = base & 127;
    const float4* gp =
        reinterpret_cast<const float4*>(A + (size_t)(m0 + r) * D_HID + c);
    float4 u0 = gp[0];
    float4 u1 = gp[1];
    float* sp = &As[r * 130 + c];
    reinterpret_cast<float2*>(sp)[0] = make_float2(u0.x, u0.y);
    reinterpret_cast<float2*>(sp)[1] = make_float2(u0.z, u0.w);
    reinterpret_cast<float2*>(sp)[2] = make_float2(u1.x, u1.y);
    reinterpret_cast<float2*>(sp)[3] = make_float2(u1.z, u1.w);
  }
  __syncthreads();

  const int lane  = tid & 31;
  const int ntile = tid >> 5;      // 0..7 -> N columns [16*ntile, 16*ntile+16)
  const int half  = lane >> 4;     // ISA A/B fragment lane-half
  const int mrow  = lane & 15;

  v8f acc = {};
  const float2* Bp =
      reinterpret_cast<const float2*>(Wp) + (size_t)ntile * 1024 + lane;

#pragma unroll
  for (int t = 0; t < 32; ++t) {
    int k = t * 4 + half * 2;
    float2 a2 = *reinterpret_cast<const float2*>(&As[mrow * 130 + k]);
    float2 b2 = Bp[t * 32];
    v2f av; av.x = a2.x; av.y = a2.y;
    v2f bv; bv.x = b2.x; bv.y = b2.y;
    acc = __builtin_amdgcn_wmma_f32_16x16x4_f32(
        /*neg_a=*/false, av, /*neg_b=*/false, bv,
        /*c_mod=*/(short)0, acc, /*reuse_a=*/false, /*reuse_b=*/false);
  }

  // Epilogue. C/D layout: acc[r] = C[m = r + 8*half][n = lane&15] (per N-tile).
  const int n = ntile * 16 + (lane & 15);
  float scale = 1.0f, shift = 0.0f;
  if (bn_g) {
    scale = bn_g[n] * rsqrtf(bn_v[n] + BN_EPS);
    shift = bn_b[n] - bn_m[n] * scale;
  }
  const float bs = bias[n];
#pragma unroll
  for (int r = 0; r < 8; ++r) {
    int m = m0 + r + half * 8;
    float v = (acc[r] + bs) * scale + shift;
    out[(size_t)m * D_HID + n] = fmaxf(v, 0.0f);
  }
}

// ---------------------------------------------------------------------------
// Zero-fill n floats.
// ---------------------------------------------------------------------------
__global__ void gin_zero(float* __restrict__ p, int n) {
  int i = blockIdx.x * blockDim.x + threadIdx.x;
  if (i < n) p[i] = 0.0f;
}

// ---------------------------------------------------------------------------
// Graph pooling: one wave32 per node, atomic adds into g[batch[node]].
// ---------------------------------------------------------------------------
__global__ void gin_pool_add(const float* __restrict__ h,
                             const int* __restrict__ batch,
                             float* __restrict__ g, int n_nodes) {
  int t = blockIdx.x * blockDim.x + threadIdx.x;
  int node = t >> 5;
  int lane = t & 31;
  if (node >= n_nodes) return;
  int gid = batch[node];
  float4 v = reinterpret_cast<const float4*>(h + (size_t)node * D_HID)[lane];
  float* o = g + (size_t)gid * D_HID + lane * 4;
  atomicAdd(o + 0, v.x);
  atomicAdd(o + 1, v.y);
  atomicAdd(o + 2, v.z);
  atomicAdd(o + 3, v.w);
}

// ---------------------------------------------------------------------------
// Head: logits = g2 @ lin2_w + lin2_b ; out = log_softmax(logits).
// One thread per graph (1000 * 1280 MACs total: negligible).
// ---------------------------------------------------------------------------
__global__ void gin_head(const float* __restrict__ g2,
                         const float* __restrict__ w,     // [128,10]
                         const float* __restrict__ b,     // [10]
                         float* __restrict__ out, int n_graphs) {
  int r = blockIdx.x * blockDim.x + threadIdx.x;
  if (r >= n_graphs) return;
  const float* row = g2 + (size_t)r * D_HID;
  float logits[N_CLASSES];
#pragma unroll
  for (int c = 0; c < N_CLASSES; ++c) {
    float acc = b[c];
    for (int k = 0; k < D_HID; ++k) acc += row[k] * w[k * N_CLASSES + c];
    logits[c] = acc;
  }
  float mx = logits[0];
#pragma unroll
  for (int c = 1; c < N_CLASSES; ++c) mx = fmaxf(mx, logits[c]);
  float sum = 0.0f;
#pragma unroll
  for (int c = 0; c < N_CLASSES; ++c) sum += expf(logits[c] - mx);
  float lse = mx + logf(sum);
#pragma unroll
  for (int c = 0; c < N_CLASSES; ++c) out[(size_t)r * N_CLASSES + c] = logits[c] - lse;
}

// ---------------------------------------------------------------------------
extern "C" void kernel_launch(void* const* d_in, const int* in_sizes, int n_in,
                              void* d_out, int out_size, void* d_ws, size_t ws_size,
                              hipStream_t stream) {
  (void)in_sizes; (void)n_in; (void)out_size; (void)ws_size;

  const float* x = (const float*)d_in[0];
  auto lp = [&](int L, int off) -> const float* {
    return (const float*)d_in[1 + 13 * L + off];
  };
  const float* lin1_w = (const float*)d_in[40];
  const float* lin1_b = (const float*)d_in[41];
  const float* lin2_w = (const float*)d_in[42];
  const float* lin2_b = (const float*)d_in[43];
  const int*   eidx   = (const int*)d_in[44];
  const int*   batch  = (const int*)d_in[45];
  const int* src = eidx;
  const int* dst = eidx + N_EDGES;

  float* ws    = (float*)d_ws;
  float* bufA  = ws;
  float* bufB  = bufA + (size_t)N_NODES * D_HID;
  float* G     = bufB + (size_t)N_NODES * D_HID;
  float* G2    = G + (size_t)GPAD * D_HID;
  float* packW = G2 + (size_t)GPAD * D_HID;       // 7 * 16384 floats

  // --- repack all weights into WMMA B-fragment order (every call; stateless)
  for (int L = 0; L < 3; ++L) {
    gin_pack_w<<<32, 256, 0, stream>>>(lp(L, 1), packW + (size_t)(2 * L) * 16384);
    gin_pack_w<<<32, 256, 0, stream>>>(lp(L, 7), packW + (size_t)(2 * L + 1) * 16384);
  }
  gin_pack_w<<<32, 256, 0, stream>>>(lin1_w, packW + (size_t)6 * 16384);

  // --- 3 GIN layers, ping-pong between bufA/bufB
  const float* hin = x;
  for (int L = 0; L < 3; ++L) {
    float* tmp = (L & 1) ? bufB : bufA;   // aggregation buffer
    float* mid = (L & 1) ? bufA : bufB;   // after first MLP linear
    float* hout = tmp;                    // tmp is dead after GEMM1 -> reuse

    gin_scale_copy<<<(N_NODES * D_HID / 4 + 255) / 256, 256, 0, stream>>>(
        hin, lp(L, 0), tmp, N_NODES * D_HID / 4);
    gin_scatter_add<<<(N_EDGES * 32) / 256, 256, 0, stream>>>(
        hin, src, dst, tmp, N_EDGES);
    gin_gemm_bn_relu<<<N_NODES / 16, 256, 0, stream>>>(
        tmp, packW + (size_t)(2 * L) * 16384,
        lp(L, 2), lp(L, 3), lp(L, 4), lp(L, 5), lp(L, 6), mid);
    gin_gemm_bn_relu<<<N_NODES / 16, 256, 0, stream>>>(
        mid, packW + (size_t)(2 * L + 1) * 16384,
        lp(L, 8), lp(L, 9), lp(L, 10), lp(L, 11), lp(L, 12), hout);
    hin = hout;
  }

  // --- global_add_pool into 16-padded [1008,128] buffer
  gin_zero<<<(GPAD * D_HID + 255) / 256, 256, 0, stream>>>(G, GPAD * D_HID);
  gin_pool_add<<<(N_NODES * 32) / 256, 256, 0, stream>>>(hin, batch, G, N_NODES);

  // --- lin1 + relu (identity BN), then lin2 + log_softmax
  gin_gemm_bn_relu<<<GPAD / 16, 256, 0, stream>>>(
      G, packW + (size_t)6 * 16384, lin1_b,
      nullptr, nullptr, nullptr, nullptr, G2);
  gin_head<<<(N_GRAPHS + 255) / 256, 256, 0, stream>>>(
      G2, lin2_w, lin2_b, (float*)d_out, N_GRAPHS);
}